// NativeSpmv_56916906606998
// MI455X (gfx1250) — compile-verified
//
#include <hip/hip_runtime.h>
#include <hip/hip_bf16.h>
#include <cstdint>

// COO SpMM, N=100000 rows, F=128 features, NNZ=3.2M edges.
// out[row[e]] += val[e] * x[col[e]]   (f32, atomically)
//
// Memory-bound (0.25 FLOP/byte): WMMA provides nothing here. We use the
// CDNA5 paths that do matter: async global->LDS staging of the edge stream
// (ASYNCcnt + s_wait_asynccnt), global_prefetch for the dependent gather,
// wave32 mapping (32 lanes x float4 == one 128-float row), and no-return
// f32 global atomics.

#define WPB   8     // waves per block (blockDim = 256, wave32)
#define CHUNK 32    // edges per chunk: one edge per lane in the async stage
#define F     128   // feature width (fixed by the reference)

typedef __attribute__((address_space(3))) const void lds_cv_t;

// 32-bit LDS byte offset of a __shared__ object (addrspacecast strips the
// generic aperture bits; AS(3) pointers are the raw LDS offsets the
// async-to-LDS instructions want in their VDST operand).
__device__ __forceinline__ unsigned ldsOffset(const void* p) {
    return (unsigned)(uintptr_t)(lds_cv_t*)p;
}

// CDNA5 async copy engine: per-lane global -> LDS, tracked by ASYNCcnt.
__device__ __forceinline__ void asyncLoadB64(unsigned ldsOff, const void* gaddr) {
    asm volatile("global_load_async_to_lds_b64 %0, %1, off"
                 :: "v"(ldsOff), "v"(gaddr) : "memory");
}
__device__ __forceinline__ void asyncLoadB32(unsigned ldsOff, const void* gaddr) {
    asm volatile("global_load_async_to_lds_b32 %0, %1, off"
                 :: "v"(ldsOff), "v"(gaddr) : "memory");
}

__global__ __launch_bounds__(256, 1) void spmm_coo_atomic(
    const float* __restrict__ x,
    const long long* __restrict__ rowIdx,
    const long long* __restrict__ colIdx,
    const float* __restrict__ val,
    float* __restrict__ out,
    long long nnz, long long nchunks)
{
    // Double-buffered per-wave edge staging: 2*8*32*(8+8+4) = 10240 B LDS.
    __shared__ long long s_row[2][WPB][CHUNK];
    __shared__ long long s_col[2][WPB][CHUNK];
    __shared__ float     s_val[2][WPB][CHUNK];

    const int lane = threadIdx.x & 31;
    const int w    = threadIdx.x >> 5;
    const long long gw     = (long long)blockIdx.x * WPB + w;
    const long long nwaves = (long long)gridDim.x * WPB;

    if (gw >= nchunks) return;

    // Stage one 32-edge chunk into LDS buffer b: 3 async instructions total.
    auto issue = [&](long long c, int b) {
        long long e = c * CHUNK + lane;
        if (e >= nnz) e = nnz - 1;                 // clamp tail (not processed)
        asyncLoadB64(ldsOffset(&s_row[b][w][lane]), rowIdx + e);
        asyncLoadB64(ldsOffset(&s_col[b][w][lane]), colIdx + e);
        asyncLoadB32(ldsOffset(&s_val[b][w][lane]), val + e);
    };

    int buf = 0;
    issue(gw, 0);

    for (long long c = gw; c < nchunks; c += nwaves) {
        const long long cn = c + nwaves;
        const bool hasNext = (cn < nchunks);
        if (hasNext) {
            issue(cn, buf ^ 1);
            // Async loads complete in order: <=3 outstanding means the 3 ops
            // of the *current* chunk have landed in LDS.
            asm volatile("s_wait_asynccnt 3" ::: "memory");
        } else {
            asm volatile("s_wait_asynccnt 0" ::: "memory");
        }

        const long long base = c * CHUNK;
        const int m = (nnz - base < CHUNK) ? (int)(nnz - base) : CHUNK;

        // Prefetch pass: hide the dependent-gather latency. Each lane group
        // (lane&3)*128B covers one cacheline of the 512 B x-row.
        for (int i = 0; i < m; ++i) {
            const long long cl = s_col[buf][w][i];
            __builtin_prefetch(x + cl * F + (lane & 3) * 32, 0, 3);
        }

        // Compute pass: 1 b128 load + 4 FMAs + 4 no-return f32 atomics per edge.
#pragma unroll 4
        for (int i = 0; i < m; ++i) {
            const long long r  = s_row[buf][w][i];   // wave-uniform LDS reads
            const long long cl = s_col[buf][w][i];
            const float     v  = s_val[buf][w][i];

            const float4 xv = *((const float4*)(x + cl * F) + lane);
            float* op = out + r * F + (lane << 2);
            __hip_atomic_fetch_add(op + 0, v * xv.x, __ATOMIC_RELAXED, __HIP_MEMORY_SCOPE_AGENT);
            __hip_atomic_fetch_add(op + 1, v * xv.y, __ATOMIC_RELAXED, __HIP_MEMORY_SCOPE_AGENT);
            __hip_atomic_fetch_add(op + 2, v * xv.z, __ATOMIC_RELAXED, __HIP_MEMORY_SCOPE_AGENT);
            __hip_atomic_fetch_add(op + 3, v * xv.w, __ATOMIC_RELAXED, __HIP_MEMORY_SCOPE_AGENT);
        }
        buf ^= 1;
    }
}

extern "C" void kernel_launch(void* const* d_in, const int* in_sizes, int n_in,
                              void* d_out, int out_size, void* d_ws, size_t ws_size,
                              hipStream_t stream) {
    (void)n_in; (void)d_ws; (void)ws_size;
    const float*     x     = (const float*)d_in[0];
    const long long* A_ind = (const long long*)d_in[1];   // int64 [2, NNZ]
    const float*     A_val = (const float*)d_in[2];

    const long long nnz = in_sizes[2];                    // NNZ
    const long long nchunks = (nnz + CHUNK - 1) / CHUNK;  // 100000 for ref sizes

    const long long* rowIdx = A_ind;          // A_ind[0, :]
    const long long* colIdx = A_ind + nnz;    // A_ind[1, :]

    // out is poisoned by the harness; atomics need a zeroed accumulator.
    hipMemsetAsync(d_out, 0, (size_t)out_size * sizeof(float), stream);

    const int blocks = 1024;                  // 8192 waves -> ~12 chunks each
    spmm_coo_atomic<<<blocks, 256, 0, stream>>>(
        x, rowIdx, colIdx, A_val, (float*)d_out, nnz, nchunks);
}